// GCN_1872605741624
// MI455X (gfx1250) — compile-verified
//
#include <hip/hip_runtime.h>
#include <hip/hip_bf16.h>

typedef float v2f __attribute__((ext_vector_type(2)));
typedef float v8f __attribute__((ext_vector_type(8)));

#define DIM 128
#define LA_STRIDE 132
#define LWT_STRIDE 132

// ---------------- degree / norm kernels ----------------

__global__ void init_deg(float* __restrict__ dout, float* __restrict__ din, int N) {
    int i = blockIdx.x * blockDim.x + threadIdx.x;
    if (i < N) { dout[i] = 1.0f; din[i] = 1.0f; }   // self-loop contributes 1 to each
}

__global__ void count_deg(const int* __restrict__ src, const int* __restrict__ dst,
                          float* __restrict__ dout, float* __restrict__ din, int E) {
    int e = blockIdx.x * blockDim.x + threadIdx.x;
    if (e < E) {
        atomicAdd(&dout[src[e]], 1.0f);
        atomicAdd(&din[dst[e]], 1.0f);
    }
}

__global__ void rsqrt_inplace(float* __restrict__ dout, float* __restrict__ din, int N) {
    int i = blockIdx.x * blockDim.x + threadIdx.x;
    if (i < N) {
        dout[i] = rsqrtf(dout[i]);   // -> norm_src
        din[i]  = rsqrtf(din[i]);    // -> norm_dst
    }
}

// ---------------- aggregation kernels ----------------

// agg[i] = h[i] * norm_src[i]  (self-loop term); one thread = float4
__global__ void init_agg(const float* __restrict__ h, const float* __restrict__ nsrc,
                         float* __restrict__ agg, int N) {
    int t = blockIdx.x * blockDim.x + threadIdx.x;
    int row = t >> 5;
    int c   = (t & 31) << 2;
    if (row < N) {
        float ns = nsrc[row];
        float4 v = *(const float4*)(h + (size_t)row * DIM + c);
        float4 o; o.x = v.x * ns; o.y = v.y * ns; o.z = v.z * ns; o.w = v.w * ns;
        *(float4*)(agg + (size_t)row * DIM + c) = o;
    }
}

// one wave32 per edge; lane handles 4 contiguous features (float4 gather + 4 f32 atomics).
// h (51.2 MB) and agg (51.2 MB) both fit in the 192 MB L2 -> runs at L2 speed.
__global__ void scatter_edges(const float* __restrict__ h, const float* __restrict__ nsrc,
                              const int* __restrict__ src, const int* __restrict__ dst,
                              float* __restrict__ agg, int E) {
    int wave = (blockIdx.x * blockDim.x + threadIdx.x) >> 5;
    int lane = threadIdx.x & 31;
    if (wave < E) {
        int s = src[wave];
        int d = dst[wave];
        float ns = nsrc[s];
        float4 v = *(const float4*)(h + (size_t)s * DIM + (lane << 2));
        float* a = agg + (size_t)d * DIM + (lane << 2);
        atomicAdd(a + 0, v.x * ns);
        atomicAdd(a + 1, v.y * ns);
        atomicAdd(a + 2, v.z * ns);
        atomicAdd(a + 3, v.w * ns);
    }
}

// ---------------- GEMM: out = act( (A * norm_dst_row) @ W + bias ) ----------------
// Block = 256 threads = 8 waves; block covers 128 rows, wave covers 16 rows.
// WMMA f32 16x16x4; K=128 -> 32 k-steps; 8 column tiles of 16.
// W is staged TRANSPOSED in LDS so the per-lane B fragment is one ds_load_b64.

template <bool RELU>
__global__ __launch_bounds__(256)
void gcn_gemm(const float* __restrict__ A,      // [N,128] aggregated messages
              const float* __restrict__ ndst,   // [N] norm_dst
              const float* __restrict__ W,      // [128,128] row-major (k,n)
              const float* __restrict__ bias,   // [128]
              float* __restrict__ out,          // [N,128]
              int N) {
    __shared__ float LWT[DIM * LWT_STRIDE];          // W transposed: LWT[n*stride + k]
    __shared__ float LA[8 * 16 * LA_STRIDE];         // per-wave padded A tiles

    const int tid  = threadIdx.x;
    const int wave = tid >> 5;
    const int lane = tid & 31;
    const int m    = lane & 15;
    const int hi   = lane >> 4;

    // cooperative load of W into LDS, transposed (one-time cost per block)
    for (int idx = tid; idx < DIM * (DIM / 4); idx += 256) {
        int r = idx >> 5;             // k
        int c = (idx & 31) << 2;      // n base
        float4 v = *(const float4*)(W + (size_t)r * DIM + c);
        LWT[(c + 0) * LWT_STRIDE + r] = v.x;
        LWT[(c + 1) * LWT_STRIDE + r] = v.y;
        LWT[(c + 2) * LWT_STRIDE + r] = v.z;
        LWT[(c + 3) * LWT_STRIDE + r] = v.w;
    }

    // stage this wave's 16x128 A tile, scaled by norm_dst (coalesced float4)
    const int blockBase = blockIdx.x * 128;
    const int rowBase   = blockBase + wave * 16;
    float* la = &LA[wave * 16 * LA_STRIDE];
    for (int idx = lane; idx < 16 * (DIM / 4); idx += 32) {
        int r = idx >> 5;
        int c = (idx & 31) << 2;
        int gr = rowBase + r;
        if (gr > N - 1) gr = N - 1;                 // clamp for tail (keeps EXEC full later)
        float s = ndst[gr];
        float4 v = *(const float4*)(A + (size_t)gr * DIM + c);
        float4 o; o.x = v.x * s; o.y = v.y * s; o.z = v.z * s; o.w = v.w * s;
        *(float4*)(&la[r * LA_STRIDE + c]) = o;
    }
    __syncthreads();

    v8f zero;
#pragma unroll
    for (int j = 0; j < 8; ++j) zero[j] = 0.0f;
    v8f acc[8];
#pragma unroll
    for (int nt = 0; nt < 8; ++nt) acc[nt] = zero;

    for (int kb = 0; kb < DIM; kb += 4) {
        const int ka = kb + 2 * hi;                  // A: lanes 0-15 K={kb,kb+1}, 16-31 K={kb+2,kb+3}
        v2f av;
        {
            float2 t = *(const float2*)&la[m * LA_STRIDE + ka];
            av.x = t.x; av.y = t.y;
        }
#pragma unroll
        for (int nt = 0; nt < 8; ++nt) {
            const int n = nt * 16 + m;
            float2 tb = *(const float2*)&LWT[n * LWT_STRIDE + ka];   // one ds_load_b64
            v2f bv; bv.x = tb.x; bv.y = tb.y;
            acc[nt] = __builtin_amdgcn_wmma_f32_16x16x4_f32(
                false, av, false, bv, (short)0, acc[nt], false, false);
        }
    }

    // C/D layout: VGPR j -> row (j + 8*hi), col = nt*16 + m
    if (blockBase + 128 <= N) {
        // fast path: whole tile in range, no per-lane predication
#pragma unroll
        for (int nt = 0; nt < 8; ++nt) {
            const int col = nt * 16 + m;
            const float bv = bias[col];
#pragma unroll
            for (int j = 0; j < 8; ++j) {
                const int grow = rowBase + j + 8 * hi;
                float v = acc[nt][j] + bv;
                if (RELU) v = fmaxf(v, 0.0f);
                out[(size_t)grow * DIM + col] = v;
            }
        }
    } else {
#pragma unroll
        for (int nt = 0; nt < 8; ++nt) {
            const int col = nt * 16 + m;
            const float bv = bias[col];
#pragma unroll
            for (int j = 0; j < 8; ++j) {
                const int grow = rowBase + j + 8 * hi;
                if (grow < N) {
                    float v = acc[nt][j] + bv;
                    if (RELU) v = fmaxf(v, 0.0f);
                    out[(size_t)grow * DIM + col] = v;
                }
            }
        }
    }
}

// ---------------- launcher ----------------

extern "C" void kernel_launch(void* const* d_in, const int* in_sizes, int n_in,
                              void* d_out, int out_size, void* d_ws, size_t ws_size,
                              hipStream_t stream) {
    const float* inputs = (const float*)d_in[0];
    const float* Ws     = (const float*)d_in[1];
    const float* bs     = (const float*)d_in[2];
    const int*   src    = (const int*)d_in[3];
    const int*   dst    = (const int*)d_in[4];
    float*       outp   = (float*)d_out;

    const int N = in_sizes[0] / DIM;
    const int L = in_sizes[1] / (DIM * DIM);
    const int E = in_sizes[3];

    float* deg_out = (float*)d_ws;                 // -> norm_src after rsqrt
    float* deg_in  = deg_out + N;                  // -> norm_dst after rsqrt
    float* bufA    = deg_in + N;                   // N*DIM
    float* bufB    = bufA + (size_t)N * DIM;       // N*DIM

    init_deg<<<(N + 255) / 256, 256, 0, stream>>>(deg_out, deg_in, N);
    count_deg<<<(E + 255) / 256, 256, 0, stream>>>(src, dst, deg_out, deg_in, E);
    rsqrt_inplace<<<(N + 255) / 256, 256, 0, stream>>>(deg_out, deg_in, N);

    const float* h = inputs;
    for (int l = 0; l < L; ++l) {
        float* agg = (l & 1) ? bufB : bufA;

        long long nthr_init = (long long)N * 32;
        init_agg<<<(int)((nthr_init + 255) / 256), 256, 0, stream>>>(h, deg_out, agg, N);

        long long nthr_sc = (long long)E * 32;
        scatter_edges<<<(int)((nthr_sc + 255) / 256), 256, 0, stream>>>(h, deg_out, src, dst, agg, E);

        float* o = (l == L - 1) ? outp : agg;      // in-place safe: wave reads its rows to LDS first
        const float* Wl = Ws + (size_t)l * DIM * DIM;
        const float* bl = bs + (size_t)l * DIM;
        if (l < L - 1) {
            gcn_gemm<true><<<(N + 127) / 128, 256, 0, stream>>>(agg, deg_in, Wl, bl, o, N);
        } else {
            gcn_gemm<false><<<(N + 127) / 128, 256, 0, stream>>>(agg, deg_in, Wl, bl, o, N);
        }
        h = o;
    }
}